// SpatialAttention_31078383354570
// MI455X (gfx1250) — compile-verified
//
#include <hip/hip_runtime.h>

typedef __attribute__((ext_vector_type(2))) float v2f;
typedef __attribute__((ext_vector_type(8))) float v8f;

#define NHEAD 8
#define HDIM  32
#define NTOK  64
#define LDQ   36   // row stride (floats) for q/k/v LDS tiles, mult of 4 for float4
#define LDP   68   // row stride (floats) for score/prob LDS tile, mult of 4
#define BT_SZ 1800 // 225 * 8 bias table entries

__launch_bounds__(128, 2)
__global__ void swin_attn_wmma_f32(const float* __restrict__ q,
                                   const float* __restrict__ k,
                                   const float* __restrict__ v,
                                   const float* __restrict__ bias_table,
                                   float* __restrict__ x_out,
                                   float* __restrict__ attn_out)
{
    __shared__ float qs[NTOK * LDQ];
    __shared__ float ks[NTOK * LDQ];
    __shared__ float vs[NTOK * LDQ];
    __shared__ float Ps[NTOK * LDP];
    __shared__ float bt[BT_SZ];

    const int tid    = threadIdx.x;
    const int wave   = tid >> 5;
    const int lane   = tid & 31;
    const int lane16 = lane & 15;
    const int khalf  = lane >> 4;          // which K-pair this half-wave owns

    const int bh = blockIdx.x;             // b*8 + h
    const int b  = bh >> 3;
    const int h  = bh & 7;

    const float scale = 0.17677669529663687f;  // 32^-0.5

    // ---- stage q,k,v head slices into LDS (q pre-scaled), bias table too ----
    const size_t base = (size_t)b * (NTOK * 256) + (size_t)h * HDIM;
    for (int idx = tid; idx < NTOK * (HDIM / 4); idx += 128) {
        int n  = idx >> 3;
        int c4 = (idx & 7) * 4;
        float4 qq = *(const float4*)(q + base + (size_t)n * 256 + c4);
        float4 kk = *(const float4*)(k + base + (size_t)n * 256 + c4);
        float4 vv = *(const float4*)(v + base + (size_t)n * 256 + c4);
        float* qd = &qs[n * LDQ + c4];
        qd[0] = qq.x * scale; qd[1] = qq.y * scale;
        qd[2] = qq.z * scale; qd[3] = qq.w * scale;
        *(float4*)&ks[n * LDQ + c4] = kk;
        *(float4*)&vs[n * LDQ + c4] = vv;
    }
    for (int idx = tid; idx < BT_SZ; idx += 128) bt[idx] = bias_table[idx];
    __syncthreads();

    // ---- GEMM1: S = (Q*scale) @ K^T ; wave owns 16 rows x 64 cols ----
    const int m0   = wave * 16;
    const int mrow = m0 + lane16;
    v8f acc0 = {}, acc1 = {}, acc2 = {}, acc3 = {};
#pragma unroll
    for (int kb = 0; kb < HDIM; kb += 4) {
        const int k0 = kb + khalf * 2;
        v2f a;
        a.x = qs[mrow * LDQ + k0];
        a.y = qs[mrow * LDQ + k0 + 1];
        v2f bf;
        bf.x = ks[(0 * 16 + lane16) * LDQ + k0];
        bf.y = ks[(0 * 16 + lane16) * LDQ + k0 + 1];
        acc0 = __builtin_amdgcn_wmma_f32_16x16x4_f32(false, a, false, bf, (short)0, acc0, false, false);
        bf.x = ks[(1 * 16 + lane16) * LDQ + k0];
        bf.y = ks[(1 * 16 + lane16) * LDQ + k0 + 1];
        acc1 = __builtin_amdgcn_wmma_f32_16x16x4_f32(false, a, false, bf, (short)0, acc1, false, false);
        bf.x = ks[(2 * 16 + lane16) * LDQ + k0];
        bf.y = ks[(2 * 16 + lane16) * LDQ + k0 + 1];
        acc2 = __builtin_amdgcn_wmma_f32_16x16x4_f32(false, a, false, bf, (short)0, acc2, false, false);
        bf.x = ks[(3 * 16 + lane16) * LDQ + k0];
        bf.y = ks[(3 * 16 + lane16) * LDQ + k0 + 1];
        acc3 = __builtin_amdgcn_wmma_f32_16x16x4_f32(false, a, false, bf, (short)0, acc3, false, false);
    }
    // D layout: VGPR j -> row m0 + j + 8*khalf, col tile*16 + lane16
#pragma unroll
    for (int j = 0; j < 8; ++j) {
        const int r = (m0 + j + 8 * khalf) * LDP + lane16;
        Ps[r + 0]  = acc0[j];
        Ps[r + 16] = acc1[j];
        Ps[r + 32] = acc2[j];
        Ps[r + 48] = acc3[j];
    }
    __syncthreads();

    // ---- softmax(S + bias) : 2 threads per row, shfl_xor(1) combine ----
    {
        const int r  = tid >> 1;
        const int hf = tid & 1;
        const int qh = r >> 3, qw = r & 7;
        float vals[32];
        float mx = -3.402823466e38f;
#pragma unroll
        for (int j = 0; j < 32; ++j) {
            const int c   = hf * 32 + j;
            const int kh2 = c >> 3, kw2 = c & 7;
            const int bidx = (qh - kh2 + 7) * 15 + (qw - kw2 + 7);
            float s = Ps[r * LDP + c] + bt[bidx * NHEAD + h];
            vals[j] = s;
            mx = fmaxf(mx, s);
        }
        mx = fmaxf(mx, __shfl_xor(mx, 1, 32));
        float sum = 0.0f;
#pragma unroll
        for (int j = 0; j < 32; ++j) { vals[j] = __expf(vals[j] - mx); sum += vals[j]; }
        sum += __shfl_xor(sum, 1, 32);
        const float inv = 1.0f / sum;
#pragma unroll
        for (int j = 0; j < 32; ++j) Ps[r * LDP + hf * 32 + j] = vals[j] * inv;
    }
    __syncthreads();

    // ---- write attn (coalesced float4): attn[b][h][r][c] ----
    {
        float* ao = attn_out + (size_t)bh * (NTOK * NTOK);
        for (int idx = tid; idx < NTOK * NTOK / 4; idx += 128) {
            const int r  = idx >> 4;
            const int c4 = (idx & 15) * 4;
            *(float4*)(ao + r * 64 + c4) = *(const float4*)&Ps[r * LDP + c4];
        }
    }

    // ---- GEMM2: X = P @ V (64x64 @ 64x32); wave owns 16 rows x 2 N-tiles ----
    v8f xa0 = {}, xa1 = {};
#pragma unroll
    for (int kb = 0; kb < NTOK; kb += 4) {
        const int k0 = kb + khalf * 2;
        v2f a;
        a.x = Ps[mrow * LDP + k0];
        a.y = Ps[mrow * LDP + k0 + 1];
        v2f bf;
        bf.x = vs[(k0    ) * LDQ + 0 * 16 + lane16];
        bf.y = vs[(k0 + 1) * LDQ + 0 * 16 + lane16];
        xa0 = __builtin_amdgcn_wmma_f32_16x16x4_f32(false, a, false, bf, (short)0, xa0, false, false);
        bf.x = vs[(k0    ) * LDQ + 1 * 16 + lane16];
        bf.y = vs[(k0 + 1) * LDQ + 1 * 16 + lane16];
        xa1 = __builtin_amdgcn_wmma_f32_16x16x4_f32(false, a, false, bf, (short)0, xa1, false, false);
    }
    // ---- write x: x[b][row][h*32 + col] ----
    float* xo = x_out + (size_t)b * (NTOK * 256) + (size_t)h * HDIM;
#pragma unroll
    for (int j = 0; j < 8; ++j) {
        const int row = m0 + j + 8 * khalf;
        xo[(size_t)row * 256 + 0 * 16 + lane16] = xa0[j];
        xo[(size_t)row * 256 + 1 * 16 + lane16] = xa1[j];
    }
}

extern "C" void kernel_launch(void* const* d_in, const int* in_sizes, int n_in,
                              void* d_out, int out_size, void* d_ws, size_t ws_size,
                              hipStream_t stream) {
    (void)in_sizes; (void)n_in; (void)out_size; (void)d_ws; (void)ws_size;
    const float* q  = (const float*)d_in[0];
    const float* k  = (const float*)d_in[1];
    const float* v  = (const float*)d_in[2];
    const float* bt = (const float*)d_in[3];

    const int B = 2048;
    float* x_out    = (float*)d_out;                       // [B,8,8,256]
    float* attn_out = x_out + (size_t)B * NTOK * 256;      // [B,8,64,64]

    dim3 grid(B * NHEAD);   // one workgroup per (batch, head)
    dim3 block(128);        // 4 wave32 waves
    swin_attn_wmma_f32<<<grid, block, 0, stream>>>(q, k, v, bt, x_out, attn_out);
}